// ConvNetLayer_40888088658326
// MI455X (gfx1250) — compile-verified
//
#include <hip/hip_runtime.h>
#include <math.h>

#define N_NODES 50000
#define N_EDGES 800000
#define MUL0    64
#define MUL1    32
#define NB      8
#define HID     64
#define WNUM    192
#define D_IN    160
#define AGG_W   384
#define HPAD    68
#define GPAD    34

// packed-weight offsets (floats) inside the weight region of d_ws
#define WOFF_W1S  0      // 64x64
#define WOFF_W1V  4096   // 32x32
#define WOFF_FW1  5120   // 8x64
#define WOFF_FW2  5632   // 64x192 (scaled by invdeg, cols>=160 also invs3)
#define WOFF_FB2  17920  // 192    (same scaling)
#define WOFF_W2S  18112  // 96x96  (x invs96)
#define WOFF_W2V  27328  // 96x32  (x invs96)
#define WOFF_WSCS 30400  // 64x96  (x 1/8)
#define WOFF_WSCV 36544  // 32x32  (x invs32)

#define INV8   0.125f
#define INVS32 0.17677669529663689f
#define INVS96 0.10206207261596575f
#define INVS3  0.57735026918962576f
#define INVDEG 0.25f

typedef __attribute__((ext_vector_type(2))) float v2f;
typedef __attribute__((ext_vector_type(8))) float v8f;

__device__ __forceinline__ v8f wmma4(v2f a, v2f b, v8f c) {
    // V_WMMA_F32_16X16X4_F32 : D = A(16x4) * B(4x16) + C, fp32 throughout
    return __builtin_amdgcn_wmma_f32_16x16x4_f32(
        false, a, false, b, (short)0, c, false, false);
}

__device__ __forceinline__ float silu_f(float x) {
    return x / (1.0f + __expf(-x));
}

__device__ __forceinline__ v2f ldv2(const float* p) { return *(const v2f*)p; }

// ------------------------------------------------ K0: zero agg accumulator
__global__ void k0_zero(float4* __restrict__ p, int n4) {
    int i = blockIdx.x * blockDim.x + threadIdx.x;
    if (i < n4) p[i] = make_float4(0.f, 0.f, 0.f, 0.f);
}

// ---------------------- Kprep: scale-fold + pair-interleave weight matrices
// Wp[(k/2)*2N + 2n + (k&1)] = W[k][n]*scale  -> B fragment = one b64 load
__device__ __forceinline__ void pack_mat(const float* __restrict__ W,
                                         float* __restrict__ Wp, int K, int N,
                                         float scale, int tid, int nth) {
    for (int i = tid; i < K * N; i += nth) {
        const int k = i / N, n = i - k * N;
        Wp[(k >> 1) * (2 * N) + 2 * n + (k & 1)] = W[i] * scale;
    }
}

__global__ void kprep(const float* __restrict__ W1s, const float* __restrict__ W1v,
                      const float* __restrict__ fw1, const float* __restrict__ fw2,
                      const float* __restrict__ fb2, const float* __restrict__ W2s,
                      const float* __restrict__ W2v, const float* __restrict__ Wscs,
                      const float* __restrict__ Wscv, float* __restrict__ pw) {
    const int tid = blockIdx.x * blockDim.x + threadIdx.x;
    const int nth = gridDim.x * blockDim.x;
    pack_mat(W1s, pw + WOFF_W1S, 64, 64, INV8, tid, nth);
    pack_mat(W1v, pw + WOFF_W1V, 32, 32, INVS32, tid, nth);
    pack_mat(fw1, pw + WOFF_FW1, 8, 64, 1.0f, tid, nth);
    for (int i = tid; i < 64 * WNUM; i += nth) {           // fc_w2, col-dep scale
        const int k = i / WNUM, n = i - k * WNUM;
        const float s = INVDEG * (n >= 160 ? INVS3 : 1.0f);
        (pw + WOFF_FW2)[(k >> 1) * (2 * WNUM) + 2 * n + (k & 1)] = fw2[i] * s;
    }
    for (int i = tid; i < WNUM; i += nth)
        (pw + WOFF_FB2)[i] = fb2[i] * INVDEG * (i >= 160 ? INVS3 : 1.0f);
    pack_mat(W2s, pw + WOFF_W2S, 96, 96, INVS96, tid, nth);
    pack_mat(W2v, pw + WOFF_W2V, 96, 32, INVS96, tid, nth);
    pack_mat(Wscs, pw + WOFF_WSCS, 64, 96, INV8, tid, nth);
    pack_mat(Wscv, pw + WOFF_WSCV, 32, 32, INVS32, tid, nth);
}

// --------------------------- K1: xs = s@W1_s/8, xv = v@W1_v/sqrt32 (scales packed)
__global__ void __launch_bounds__(256)
k1_node_pre(const float* __restrict__ nf, const float* __restrict__ pw,
            float* __restrict__ xs, float* __restrict__ xv) {
    const int lane = threadIdx.x & 31;
    const int wid  = threadIdx.x >> 5;
    const int half = lane >> 4;
    const int lcol = lane & 15;
    const int tile = blockIdx.x * 8 + wid;
    if (tile * 16 >= N_NODES) return;
    const int row0 = tile * 16;
    const float* w1s_p = pw + WOFF_W1S;
    const float* w1v_p = pw + WOFF_W1V;

    // xs: M=16, K=64, Ncols=64
    {
        v8f acc[4] = {};
#pragma unroll
        for (int kk = 0; kk < 16; ++kk) {
            const int k0 = kk * 4 + 2 * half;
            const v2f a = ldv2(nf + (row0 + lcol) * D_IN + k0);
#pragma unroll
            for (int ct = 0; ct < 4; ++ct) {
                const v2f b = ldv2(w1s_p + (kk * 2 + half) * 128 + 2 * (ct * 16 + lcol));
                acc[ct] = wmma4(a, b, acc[ct]);
            }
        }
#pragma unroll
        for (int ct = 0; ct < 4; ++ct)
#pragma unroll
            for (int r = 0; r < 8; ++r)
                xs[(row0 + r + 8 * half) * MUL0 + ct * 16 + lcol] = acc[ct][r];
    }

    // xv: per component, M=16, K=32, Ncols=32
#pragma unroll
    for (int c = 0; c < 3; ++c) {
        v8f acc[2] = {};
#pragma unroll
        for (int kk = 0; kk < 8; ++kk) {
            const int k0 = kk * 4 + 2 * half;
            v2f a;
            a.x = nf[(row0 + lcol) * D_IN + MUL0 + 3 * k0 + c];
            a.y = nf[(row0 + lcol) * D_IN + MUL0 + 3 * (k0 + 1) + c];
#pragma unroll
            for (int ct = 0; ct < 2; ++ct) {
                const v2f b = ldv2(w1v_p + (kk * 2 + half) * 64 + 2 * (ct * 16 + lcol));
                acc[ct] = wmma4(a, b, acc[ct]);
            }
        }
#pragma unroll
        for (int ct = 0; ct < 2; ++ct)
#pragma unroll
            for (int r = 0; r < 8; ++r)
                xv[(row0 + r + 8 * half) * 96 + 3 * (ct * 16 + lcol) + c] = acc[ct][r];
    }
}

// layer-2 MLP tile: 16 edges x 16 cols, K=64, A from LDS hidden, B/bias pre-scaled
__device__ __forceinline__ v8f mlp2_tile(const float* hbase,
                                         const float* __restrict__ fw2_p,
                                         const float* __restrict__ fb2_s,
                                         int ct, int lcol, int half) {
    v8f acc;
    const float b2 = fb2_s[ct * 16 + lcol];
#pragma unroll
    for (int r = 0; r < 8; ++r) acc[r] = b2;
#pragma unroll
    for (int kk = 0; kk < 16; ++kk) {
        const int k0 = kk * 4 + 2 * half;
        const v2f a = ldv2(hbase + lcol * HPAD + k0);
        const v2f b = ldv2(fw2_p + (kk * 2 + half) * (2 * WNUM) + 2 * (ct * 16 + lcol));
        acc = wmma4(a, b, acc);
    }
    return acc;
}

// -------------------- K2: edge MLP (WMMA) + tensor-product messages + scatter
__global__ void __launch_bounds__(256)
k2_edge(const float* __restrict__ edge_sh, const float* __restrict__ edge_embed,
        const float* __restrict__ fc_b1, const float* __restrict__ pw,
        const int* __restrict__ edge_index, const float* __restrict__ xs,
        const float* __restrict__ xv, float* __restrict__ agg) {
    __shared__ float lds_h[8][16][HPAD];
    const int lane = threadIdx.x & 31;
    const int wid  = threadIdx.x >> 5;
    const int half = lane >> 4;
    const int lcol = lane & 15;
    const int eb   = (blockIdx.x * 8 + wid) * 16;
    const float* fw1_p = pw + WOFF_FW1;
    const float* fw2_p = pw + WOFF_FW2;
    const float* fb2_s = pw + WOFF_FB2;

    // layer 1: hidden = silu(edge_embed @ fc_w1 + b1), K=8, Ncols=64
#pragma unroll
    for (int ct = 0; ct < 4; ++ct) {
        v8f acc;
        const float b1 = fc_b1[ct * 16 + lcol];
#pragma unroll
        for (int r = 0; r < 8; ++r) acc[r] = b1;
#pragma unroll
        for (int kk = 0; kk < 2; ++kk) {
            const int k0 = kk * 4 + 2 * half;
            const v2f a = ldv2(edge_embed + (eb + lcol) * NB + k0);
            const v2f b = ldv2(fw1_p + (kk * 2 + half) * 128 + 2 * (ct * 16 + lcol));
            acc = wmma4(a, b, acc);
        }
#pragma unroll
        for (int r = 0; r < 8; ++r)
            lds_h[wid][r + 8 * half][ct * 16 + lcol] = silu_f(acc[r]);
    }
    __syncthreads();

    // per-edge metadata for this lane's 8 rows
    int srcs[8], dsts[8];
    float s0v[8], s1v[8][3];
#pragma unroll
    for (int r = 0; r < 8; ++r) {
        const int e = eb + r + 8 * half;
        srcs[r] = edge_index[e];
        dsts[r] = edge_index[N_EDGES + e];
        const float4 sh = *(const float4*)(edge_sh + e * 4);
        s0v[r] = sh.x;
        s1v[r][0] = sh.y;
        s1v[r][1] = sh.z;
        s1v[r][2] = sh.w;
    }
    const float* hbase = &lds_h[wid][0][0];

    // tiles 0..3: w_ss -> m0_a  (agg cols 0..63); invdeg folded into w
#pragma unroll
    for (int ct = 0; ct < 4; ++ct) {
        v8f acc = mlp2_tile(hbase, fw2_p, fb2_s, ct, lcol, half);
        const int j = ct * 16 + lcol;
#pragma unroll
        for (int r = 0; r < 8; ++r) {
            const float xse = xs[srcs[r] * MUL0 + j];
            atomicAdd(&agg[dsts[r] * AGG_W + j], acc[r] * xse * s0v[r]);
        }
    }
    // tiles 4..7: w_sv -> m1_a  (agg cols 96 + 3j + c)
#pragma unroll
    for (int ct = 4; ct < 8; ++ct) {
        v8f acc = mlp2_tile(hbase, fw2_p, fb2_s, ct, lcol, half);
        const int j = ct * 16 + lcol - MUL0;
#pragma unroll
        for (int r = 0; r < 8; ++r) {
            const float t = acc[r] * xs[srcs[r] * MUL0 + j];
            float* base = &agg[dsts[r] * AGG_W + 96 + 3 * j];
            atomicAdd(base + 0, t * s1v[r][0]);
            atomicAdd(base + 1, t * s1v[r][1]);
            atomicAdd(base + 2, t * s1v[r][2]);
        }
    }
    // tiles 8..9: w_vs -> m1_b  (agg cols 288 + 3k + c)
#pragma unroll
    for (int ct = 8; ct < 10; ++ct) {
        v8f acc = mlp2_tile(hbase, fw2_p, fb2_s, ct, lcol, half);
        const int k = ct * 16 + lcol - 128;
#pragma unroll
        for (int r = 0; r < 8; ++r) {
            const float t = acc[r] * s0v[r];
            const float* xvp = &xv[srcs[r] * 96 + 3 * k];
            float* base = &agg[dsts[r] * AGG_W + 288 + 3 * k];
            atomicAdd(base + 0, t * xvp[0]);
            atomicAdd(base + 1, t * xvp[1]);
            atomicAdd(base + 2, t * xvp[2]);
        }
    }
    // tiles 10..11: w_vv -> m0_b (agg cols 64 + k); invdeg*invs3 folded into w
#pragma unroll
    for (int ct = 10; ct < 12; ++ct) {
        v8f acc = mlp2_tile(hbase, fw2_p, fb2_s, ct, lcol, half);
        const int k = ct * 16 + lcol - 160;
#pragma unroll
        for (int r = 0; r < 8; ++r) {
            const float* xvp = &xv[srcs[r] * 96 + 3 * k];
            const float d = xvp[0] * s1v[r][0] + xvp[1] * s1v[r][1] + xvp[2] * s1v[r][2];
            atomicAdd(&agg[dsts[r] * AGG_W + 64 + k], acc[r] * d);
        }
    }
}

// ------------------- K3: node post GEMMs + skip connections + SiLU gating
__global__ void __launch_bounds__(256)
k3_out(const float* __restrict__ nf, const float* __restrict__ agg,
       const float* __restrict__ pw, float* __restrict__ out) {
    __shared__ float lds_g[8][16][GPAD];
    const int lane = threadIdx.x & 31;
    const int wid  = threadIdx.x >> 5;
    const int half = lane >> 4;
    const int lcol = lane & 15;
    const int tile = blockIdx.x * 8 + wid;
    const bool active = (tile * 16 < N_NODES);
    const int row0 = tile * 16;
    const float* w2s_p  = pw + WOFF_W2S;
    const float* w2v_p  = pw + WOFF_W2V;
    const float* wscs_p = pw + WOFF_WSCS;
    const float* wscv_p = pw + WOFF_WSCV;

    if (active) {
        // out_s = agg_s @ W2_s/sqrt96 + s @ Wsc_s/8 ; Ncols = 96 (6 tiles)
#pragma unroll
        for (int ct = 0; ct < 6; ++ct) {
            v8f acc = {};
#pragma unroll
            for (int kk = 0; kk < 24; ++kk) {       // K=96 over agg_s
                const int k0 = kk * 4 + 2 * half;
                const v2f a = ldv2(agg + (row0 + lcol) * AGG_W + k0);
                const v2f b = ldv2(w2s_p + (kk * 2 + half) * 192 + 2 * (ct * 16 + lcol));
                acc = wmma4(a, b, acc);
            }
#pragma unroll
            for (int kk = 0; kk < 16; ++kk) {       // K=64 skip (s)
                const int k0 = kk * 4 + 2 * half;
                const v2f a = ldv2(nf + (row0 + lcol) * D_IN + k0);
                const v2f b = ldv2(wscs_p + (kk * 2 + half) * 192 + 2 * (ct * 16 + lcol));
                acc = wmma4(a, b, acc);
            }
            if (ct < 4) {
#pragma unroll
                for (int r = 0; r < 8; ++r) {
                    const int row = r + 8 * half, col = ct * 16 + lcol;
                    out[(row0 + row) * D_IN + col] =
                        nf[(row0 + row) * D_IN + col] + silu_f(acc[r]);
                }
            } else {
#pragma unroll
                for (int r = 0; r < 8; ++r)
                    lds_g[wid][r + 8 * half][(ct - 4) * 16 + lcol] = silu_f(acc[r]);
            }
        }
    }
    __syncthreads();
    if (!active) return;

    // out_v = agg_v @ W2_v/sqrt96 + v @ Wsc_v/sqrt32 ; gated output
#pragma unroll
    for (int c = 0; c < 3; ++c) {
#pragma unroll
        for (int ct = 0; ct < 2; ++ct) {
            v8f acc = {};
#pragma unroll
            for (int kk = 0; kk < 24; ++kk) {       // K=96 over agg_v[:,c]
                const int k0 = kk * 4 + 2 * half;
                v2f a;
                a.x = agg[(row0 + lcol) * AGG_W + 96 + 3 * k0 + c];
                a.y = agg[(row0 + lcol) * AGG_W + 96 + 3 * (k0 + 1) + c];
                const v2f b = ldv2(w2v_p + (kk * 2 + half) * 64 + 2 * (ct * 16 + lcol));
                acc = wmma4(a, b, acc);
            }
#pragma unroll
            for (int kk = 0; kk < 8; ++kk) {        // K=32 skip (v)
                const int k0 = kk * 4 + 2 * half;
                v2f a;
                a.x = nf[(row0 + lcol) * D_IN + MUL0 + 3 * k0 + c];
                a.y = nf[(row0 + lcol) * D_IN + MUL0 + 3 * (k0 + 1) + c];
                const v2f b = ldv2(wscv_p + (kk * 2 + half) * 64 + 2 * (ct * 16 + lcol));
                acc = wmma4(a, b, acc);
            }
#pragma unroll
            for (int r = 0; r < 8; ++r) {
                const int row = r + 8 * half, k = ct * 16 + lcol;
                const float g = lds_g[wid][row][k];
                const int off = (row0 + row) * D_IN + MUL0 + 3 * k + c;
                out[off] = nf[off] + acc[r] * g;
            }
        }
    }
}

extern "C" void kernel_launch(void* const* d_in, const int* in_sizes, int n_in,
                              void* d_out, int out_size, void* d_ws, size_t ws_size,
                              hipStream_t stream) {
    const float* nf         = (const float*)d_in[0];
    const float* edge_sh    = (const float*)d_in[1];
    const float* edge_embed = (const float*)d_in[2];
    const float* W1s        = (const float*)d_in[3];
    const float* W1v        = (const float*)d_in[4];
    const float* fcw1       = (const float*)d_in[5];
    const float* fcb1       = (const float*)d_in[6];
    const float* fcw2       = (const float*)d_in[7];
    const float* fcb2       = (const float*)d_in[8];
    const float* W2s        = (const float*)d_in[9];
    const float* W2v        = (const float*)d_in[10];
    const float* Wscs       = (const float*)d_in[11];
    const float* Wscv       = (const float*)d_in[12];
    const int*   edge_index = (const int*)d_in[13];
    float* out = (float*)d_out;

    char* ws = (char*)d_ws;
    float* xs  = (float*)ws;                                       // N*64  f32
    float* xv  = (float*)(ws + (size_t)N_NODES * MUL0 * 4);        // N*96  f32
    float* agg = (float*)(ws + (size_t)N_NODES * (MUL0 + 96) * 4); // N*384 f32
    float* pw  = (float*)(ws + (size_t)N_NODES * (MUL0 + 96 + AGG_W) * 4); // weights

    kprep<<<48, 256, 0, stream>>>(W1s, W1v, fcw1, fcw2, fcb2, W2s, W2v, Wscs, Wscv, pw);

    const int n4 = N_NODES * AGG_W / 4;
    k0_zero<<<(n4 + 255) / 256, 256, 0, stream>>>((float4*)agg, n4);

    const int ntiles = N_NODES / 16;                  // 3125
    k1_node_pre<<<(ntiles + 7) / 8, 256, 0, stream>>>(nf, pw, xs, xv);

    k2_edge<<<N_EDGES / 16 / 8, 256, 0, stream>>>(edge_sh, edge_embed, fcb1, pw,
                                                  edge_index, xs, xv, agg);

    k3_out<<<(ntiles + 7) / 8, 256, 0, stream>>>(nf, agg, pw, out);
}